// Omnidirectional_3D_Mamba_41721312313756
// MI455X (gfx1250) — compile-verified
//
#include <hip/hip_runtime.h>
#include <hip/hip_bf16.h>

typedef __attribute__((ext_vector_type(2))) float v2f;
typedef __attribute__((ext_vector_type(8))) float v8f;

// Problem constants
#define NC        16          // C
#define D_INNER   64
#define NHEADS    16
#define HEADDIM   4
#define D_STATE   64
#define CONV_DIM  192
#define D_PROJ    272         // 2*64 + 2*64 + 16
#define BSZ       8
#define DD        32
#define HH        16
#define WW        16
#define LSEQ      8192        // 32*16*16
#define NPOS      65536       // BSZ*LSEQ (per dir)
#define NROWS     196608      // 3*NPOS
#define STRD      208         // packed scan stream row: dA[16] dtx[64] B[64] C[64]
#define POOLN     256

__device__ __forceinline__ float siluf(float x) { return x / (1.0f + __expf(-x)); }
__device__ __forceinline__ float softplusf(float x) {
    return (x > 20.0f) ? x : log1pf(__expf(x));
}

// ---------------------------------------------------------------------------
// Stage 1: in_proj GEMM.  rows = (dir,b,l) in the dir's sequence order.
// out[row][p] = sum_c u[row][c] * Win[dir][p][c],  K=16 via 4 chained
// V_WMMA_F32_16X16X4_F32.  Directional gather folded into A-operand load.
// ---------------------------------------------------------------------------
__global__ __launch_bounds__(256) void k_inproj(const float* __restrict__ x,
                                                const float* __restrict__ Win,
                                                float* __restrict__ zxb) {
    const int lane = threadIdx.x & 31;
    const int wv   = threadIdx.x >> 5;
    const long mtile = (long)blockIdx.x * 8 + wv;        // 12288 tiles
    const long row0  = mtile * 16;
    const int  dir   = (int)(row0 >> 16);                // /65536
    const int  M     = lane & 15;
    const int  kh    = lane >> 4;                        // 0/1 -> K offset {0,1}/{2,3}

    // row handled by this lane for the A operand
    const long r  = row0 + M;
    const long pd = r & 65535;                           // pos within dir
    const int  b  = (int)(pd >> 13);
    const int  l  = (int)(pd & 8191);
    int sp;
    if (dir == 0) {                                      // fD: l = d*256 + h*16 + w
        sp = l;
    } else if (dir == 1) {                               // fH: l = h*512 + w*32 + d
        int d = l & 31, w = (l >> 5) & 15, h = l >> 9;
        sp = d * 256 + h * 16 + w;
    } else {                                             // fW: l = w*512 + d*16 + h
        int h = l & 15, d = (l >> 4) & 31, w = l >> 9;
        sp = d * 256 + h * 16 + w;
    }
    const float* xb = x + ((long)b * NC) * (long)LSEQ + sp;

    v2f A[4];
#pragma unroll
    for (int kb = 0; kb < 4; ++kb) {                     // K = kb*4 .. kb*4+3
        int c0 = kb * 4 + kh * 2;
        A[kb].x = xb[(long)c0 * LSEQ];
        A[kb].y = xb[(long)(c0 + 1) * LSEQ];
    }

    const float* Wd   = Win + dir * D_PROJ * NC;
    float*       outp = zxb + row0 * D_PROJ;

#pragma unroll 1
    for (int nt = 0; nt < 17; ++nt) {                    // 17 * 16 = 272 cols
        const int n = nt * 16 + M;
        v8f c = {0.f, 0.f, 0.f, 0.f, 0.f, 0.f, 0.f, 0.f};
#pragma unroll
        for (int kb = 0; kb < 4; ++kb) {
            int k0 = kb * 4 + kh * 2;
            v2f bm;
            bm.x = Wd[n * NC + k0];
            bm.y = Wd[n * NC + k0 + 1];
            c = __builtin_amdgcn_wmma_f32_16x16x4_f32(false, A[kb], false, bm,
                                                      (short)0, c, false, false);
        }
#pragma unroll
        for (int j = 0; j < 8; ++j) {
            int mr = j + kh * 8;                         // D-layout: M = j + (lane/16)*8
            outp[(long)mr * D_PROJ + n] = c[j];
        }
    }
}

// ---------------------------------------------------------------------------
// Stage 2: causal depthwise conv(4) + SiLU on xBC; softplus(dt); pack the
// scan stream [dA(16) | dtx(64) | B(64) | C(64)] and store xh.
// ---------------------------------------------------------------------------
__global__ __launch_bounds__(192) void k_conv_pack(const float* __restrict__ zxb,
                                                   const float* __restrict__ conv_w,
                                                   const float* __restrict__ conv_b,
                                                   const float* __restrict__ dt_bias,
                                                   const float* __restrict__ A_log,
                                                   float* __restrict__ stream,
                                                   float* __restrict__ xh) {
    const long pos = blockIdx.x;                         // 0..196607
    const int  ch  = threadIdx.x;                        // 0..191
    const int  dir = (int)(pos >> 16);
    const int  l   = (int)(pos & 8191);

    const float* base = zxb + pos * D_PROJ + D_INNER;    // xBC pre-conv
    const float* cw   = conv_w + (dir * CONV_DIM + ch) * 4;
    float acc = conv_b[dir * CONV_DIM + ch];
#pragma unroll
    for (int k = 0; k < 4; ++k) {
        int ll = l - 3 + k;
        float v = (ll >= 0) ? base[(long)(ll - l) * D_PROJ + ch] : 0.0f;
        acc += v * cw[k];
    }
    const float sv = siluf(acc);

    float* sp = stream + pos * STRD;
    if (ch < D_INNER) {
        const int h  = ch >> 2;
        float dt = softplusf(zxb[pos * D_PROJ + 256 + h] + dt_bias[dir * NHEADS + h]);
        xh[pos * D_INNER + ch] = sv;
        sp[16 + ch] = dt * sv;                           // dtx
        if (ch < NHEADS) {
            float dt2 = softplusf(zxb[pos * D_PROJ + 256 + ch] + dt_bias[dir * NHEADS + ch]);
            sp[ch] = __expf(-dt2 * __expf(A_log[dir * NHEADS + ch]));   // dA
        }
    } else if (ch < D_INNER + D_STATE) {
        sp[80 + (ch - D_INNER)] = sv;                    // B
    } else {
        sp[144 + (ch - D_INNER - D_STATE)] = sv;         // C
    }
}

// ---------------------------------------------------------------------------
// Stage 3: selective scan.  One block per (dir,b), one wave per head.
// Lane t: p = t/8, owns n = (t%8)*8 .. +7  -> 8 state regs.
// ---------------------------------------------------------------------------
__global__ __launch_bounds__(512) void k_scan(const float* __restrict__ stream,
                                              float* __restrict__ ybuf) {
    const int sb   = blockIdx.x;                         // 0..23 = dir*8+b
    const int h    = threadIdx.x >> 5;
    const int lane = threadIdx.x & 31;
    const int p    = lane >> 3;
    const int g    = lane & 7;
    const int n0   = g * 8;

    float S[8];
#pragma unroll
    for (int i = 0; i < 8; ++i) S[i] = 0.0f;

    const float* sp = stream + (long)sb * LSEQ * STRD;
    float*       yp = ybuf + (long)sb * LSEQ * D_INNER + h * HEADDIM + p;

    for (int l = 0; l < LSEQ; ++l) {
        const float* s = sp + (long)l * STRD;
        if ((l & 7) == 0) __builtin_prefetch(s + 16 * STRD, 0, 1);  // global_prefetch_b8

        const float dA  = s[h];
        const float dtx = s[16 + h * HEADDIM + p];
        const float4 b0 = *(const float4*)(s + 80 + n0);
        const float4 b1 = *(const float4*)(s + 80 + n0 + 4);
        const float4 c0 = *(const float4*)(s + 144 + n0);
        const float4 c1 = *(const float4*)(s + 144 + n0 + 4);

        S[0] = S[0] * dA + dtx * b0.x;  S[1] = S[1] * dA + dtx * b0.y;
        S[2] = S[2] * dA + dtx * b0.z;  S[3] = S[3] * dA + dtx * b0.w;
        S[4] = S[4] * dA + dtx * b1.x;  S[5] = S[5] * dA + dtx * b1.y;
        S[6] = S[6] * dA + dtx * b1.z;  S[7] = S[7] * dA + dtx * b1.w;

        float acc = S[0] * c0.x + S[1] * c0.y + S[2] * c0.z + S[3] * c0.w
                  + S[4] * c1.x + S[5] * c1.y + S[6] * c1.z + S[7] * c1.w;
        acc += __shfl_xor(acc, 1, 32);
        acc += __shfl_xor(acc, 2, 32);
        acc += __shfl_xor(acc, 4, 32);
        if (g == 0) yp[(long)l * D_INNER] = acc;
    }
}

// ---------------------------------------------------------------------------
// Stage 4: y = ys + Dp*xh; gate with silu(z); RMSNorm(64); out_proj GEMM
// (K=64, N=16) via 16 chained fp32 WMMAs; ReLU; BN partial sums per wave.
// ---------------------------------------------------------------------------
__global__ __launch_bounds__(256) void k_outproj(const float* __restrict__ zxb,
                                                 const float* __restrict__ ybuf,
                                                 const float* __restrict__ xh,
                                                 const float* __restrict__ Dp,
                                                 const float* __restrict__ norm_w,
                                                 const float* __restrict__ Wout,
                                                 float* __restrict__ yout,
                                                 float* __restrict__ partial) {
    __shared__ float tile[8][16][68];                    // +4 pad vs 64 banks
    const int  lane = threadIdx.x & 31;
    const int  wv   = threadIdx.x >> 5;
    const long wave = (long)blockIdx.x * 8 + wv;         // 12288 waves
    const long row0 = wave * 16;
    const int  dir  = (int)(row0 >> 16);
    const int  M    = lane & 15;
    const int  kh   = lane >> 4;                         // half of K per row

    const long r = row0 + M;
    const float* yb = ybuf + r * D_INNER + kh * 32;
    const float* xb = xh   + r * D_INNER + kh * 32;
    const float* zb = zxb  + r * D_PROJ  + kh * 32;      // z = cols 0..63

    float gv[32];
    float ss = 0.0f;
#pragma unroll
    for (int i = 0; i < 32; ++i) {
        const int k = kh * 32 + i;
        const float dp = Dp[dir * NHEADS + (k >> 2)];
        const float v  = (yb[i] + dp * xb[i]) * siluf(zb[i]);
        gv[i] = v;
        ss += v * v;
    }
    ss += __shfl_xor(ss, 16, 32);
    const float rms = rsqrtf(ss * (1.0f / 64.0f) + 1e-5f);
#pragma unroll
    for (int i = 0; i < 32; ++i) {
        const int k = kh * 32 + i;
        tile[wv][M][k] = gv[i] * rms * norm_w[dir * D_INNER + k];
    }
    __syncthreads();

    const float* Wd = Wout + dir * NC * D_INNER;
    const int koff = kh * 2;
    v8f c = {0.f, 0.f, 0.f, 0.f, 0.f, 0.f, 0.f, 0.f};
#pragma unroll
    for (int kb = 0; kb < 16; ++kb) {                    // K = 64
        v2f a, bm;
        a.x  = tile[wv][M][kb * 4 + koff];
        a.y  = tile[wv][M][kb * 4 + koff + 1];
        bm.x = Wd[M * D_INNER + kb * 4 + koff];          // B[k][n=c] = Wout[c][k]
        bm.y = Wd[M * D_INNER + kb * 4 + koff + 1];
        c = __builtin_amdgcn_wmma_f32_16x16x4_f32(false, a, false, bm,
                                                  (short)0, c, false, false);
    }

    float s1 = 0.0f, s2 = 0.0f;
#pragma unroll
    for (int j = 0; j < 8; ++j) {
        const long rr = row0 + j + kh * 8;
        const float v = fmaxf(c[j], 0.0f);               // ReLU folded here
        yout[rr * NC + M] = v;
        s1 += v;
        s2 += v * v;
    }
    s1 += __shfl_xor(s1, 16, 32);
    s2 += __shfl_xor(s2, 16, 32);
    if (lane < 16) {
        partial[wave * 32 + lane]      = s1;             // per-channel sum
        partial[wave * 32 + 16 + lane] = s2;             // per-channel sumsq
    }
}

// ---------------------------------------------------------------------------
// Stage 5a: deterministic reduction of BN partials (4096 waves per dir).
// ---------------------------------------------------------------------------
__global__ __launch_bounds__(256) void k_redstats(const float* __restrict__ partial,
                                                  float* __restrict__ stats) {
    const int dir = blockIdx.x;
    const int t   = threadIdx.x;
    const int c   = t & 31;
    const int grp = t >> 5;                              // 8 groups
    float acc = 0.0f;
    for (long w = (long)dir * 4096 + grp; w < (long)(dir + 1) * 4096; w += 8)
        acc += partial[w * 32 + c];
    __shared__ float red[256];
    red[t] = acc;
    __syncthreads();
    if (grp == 0) {
        float s = red[c] + red[32 + c] + red[64 + c] + red[96 + c]
                + red[128 + c] + red[160 + c] + red[192 + c] + red[224 + c];
        stats[dir * 32 + c] = s;
    }
}

// ---------------------------------------------------------------------------
// Stage 5b: fold BN + Linear into per-row scalar, wave-wide maxpool over 32.
// One wave per (dir, b, pool).
// ---------------------------------------------------------------------------
__global__ __launch_bounds__(256) void k_pool(const float* __restrict__ yout,
                                              const float* __restrict__ stats,
                                              const float* __restrict__ gamma,
                                              const float* __restrict__ beta,
                                              const float* __restrict__ lin_w,
                                              const float* __restrict__ lin_b,
                                              float* __restrict__ out) {
    const int lane = threadIdx.x & 31;
    const int wv   = threadIdx.x >> 5;
    const int W    = blockIdx.x * 8 + wv;                // 6144 waves
    const int dir  = W >> 11;
    const int b    = (W >> 8) & 7;
    const int pool = W & 255;

    const long row = ((long)(dir * 8 + b)) * LSEQ + pool * 32 + lane;
    const float* v = yout + row * NC;

    float s = lin_b[dir];
#pragma unroll
    for (int c = 0; c < NC; ++c) {
        const float S1 = stats[dir * 32 + c];
        const float S2 = stats[dir * 32 + 16 + c];
        const float mu  = S1 * (1.0f / 65536.0f);
        const float var = S2 * (1.0f / 65536.0f) - mu * mu;
        const float rs  = rsqrtf(var + 1e-5f);
        const float gw  = gamma[dir * NC + c] * rs * lin_w[dir * NC + c];
        s += (v[c] - mu) * gw + beta[dir * NC + c] * lin_w[dir * NC + c];
    }
    s = fmaxf(s, __shfl_xor(s, 1, 32));
    s = fmaxf(s, __shfl_xor(s, 2, 32));
    s = fmaxf(s, __shfl_xor(s, 4, 32));
    s = fmaxf(s, __shfl_xor(s, 8, 32));
    s = fmaxf(s, __shfl_xor(s, 16, 32));
    if (lane == 0) out[b * 768 + dir * 256 + pool] = s;
}

// ---------------------------------------------------------------------------
extern "C" void kernel_launch(void* const* d_in, const int* in_sizes, int n_in,
                              void* d_out, int out_size, void* d_ws, size_t ws_size,
                              hipStream_t stream) {
    const float* x       = (const float*)d_in[0];
    const float* Win     = (const float*)d_in[1];
    const float* conv_w  = (const float*)d_in[2];
    const float* conv_b  = (const float*)d_in[3];
    const float* dt_bias = (const float*)d_in[4];
    const float* A_log   = (const float*)d_in[5];
    const float* Dp      = (const float*)d_in[6];
    const float* norm_w  = (const float*)d_in[7];
    const float* Wout    = (const float*)d_in[8];
    const float* gamma   = (const float*)d_in[9];
    const float* beta    = (const float*)d_in[10];
    const float* lin_w   = (const float*)d_in[11];
    const float* lin_b   = (const float*)d_in[12];
    float* out = (float*)d_out;

    float* ws = (float*)d_ws;
    float* zxb     = ws;                                 // 196608*272
    float* strm    = zxb  + (long)NROWS * D_PROJ;        // 196608*208
    float* xh      = strm + (long)NROWS * STRD;          // 196608*64
    float* ybuf    = xh   + (long)NROWS * D_INNER;       // 196608*64
    float* yout    = ybuf + (long)NROWS * D_INNER;       // 196608*16
    float* partial = yout + (long)NROWS * NC;            // 12288*32
    float* stats   = partial + (long)12288 * 32;         // 3*32

    k_inproj   <<<dim3(1536),   dim3(256), 0, stream>>>(x, Win, zxb);
    k_conv_pack<<<dim3(NROWS),  dim3(192), 0, stream>>>(zxb, conv_w, conv_b,
                                                        dt_bias, A_log, strm, xh);
    k_scan     <<<dim3(24),     dim3(512), 0, stream>>>(strm, ybuf);
    k_outproj  <<<dim3(1536),   dim3(256), 0, stream>>>(zxb, ybuf, xh, Dp, norm_w,
                                                        Wout, yout, partial);
    k_redstats <<<dim3(3),      dim3(256), 0, stream>>>(partial, stats);
    k_pool     <<<dim3(768),    dim3(256), 0, stream>>>(yout, stats, gamma, beta,
                                                        lin_w, lin_b, out);
}